// MHGCN_89318139888058
// MI455X (gfx1250) — compile-verified
//
#include <hip/hip_runtime.h>
#include <math.h>

typedef __attribute__((ext_vector_type(2))) float v2f;
typedef __attribute__((ext_vector_type(8))) float v8f;

#define B_    32
#define C_    5
#define N_    62
#define NP_   64            // padded node dim (zero rows 62,63)
#define T_    1024
#define L_    4
#define TOPK_ 49
#define NCO_  25            // (L+1)*C
#define NT_   (N_ * T_)
#define NTP_  (NP_ * T_)
#define LDSP_ 68            // LDS pitch for adj tile (conflict-free, 8B-aligned)

// ---------------------------------------------------------------------------
// Kernel 0: copy x into output channel block 0 ([B, 25, N, T]) via float4
// ---------------------------------------------------------------------------
__global__ __launch_bounds__(256)
void mh_copy_x(const float4* __restrict__ x, float* __restrict__ out) {
  size_t idx = (size_t)blockIdx.x * blockDim.x + threadIdx.x;   // float4 units
  const size_t tot = (size_t)B_ * C_ * NT_ / 4;
  if (idx >= tot) return;
  size_t b   = idx / ((size_t)C_ * NT_ / 4);
  size_t rem = idx % ((size_t)C_ * NT_ / 4);
  ((float4*)(out + b * ((size_t)NCO_ * NT_)))[rem] = x[idx];
}

// ---------------------------------------------------------------------------
// Kernel A1: 1x1 channel conv into padded workspace x1p[b,c,64,T] (float4).
// Rows n in [62,64) are written as zeros (enables unguarded WMMA loads).
// ---------------------------------------------------------------------------
__global__ __launch_bounds__(256)
void mh_conv1(const float* __restrict__ in, int chIn,
              const float* __restrict__ w, const float* __restrict__ bias,
              float* __restrict__ x1p) {
  size_t idx = (size_t)blockIdx.x * blockDim.x + threadIdx.x;   // over B*64*(T/4)
  const size_t tot = (size_t)B_ * NP_ * (T_ / 4);
  if (idx >= tot) return;
  const int t4 = (int)(idx % (T_ / 4));
  const int n  = (int)((idx / (T_ / 4)) % NP_);
  const int b  = (int)(idx / ((size_t)NP_ * (T_ / 4)));

  if (n >= N_) {                       // zero pad rows
    float4 z = make_float4(0.f, 0.f, 0.f, 0.f);
#pragma unroll
    for (int o = 0; o < C_; ++o)
      ((float4*)(x1p + ((size_t)(b * C_ + o) * NP_ + n) * T_))[t4] = z;
    return;
  }

  float4 xin[C_];
#pragma unroll
  for (int c = 0; c < C_; ++c)
    xin[c] = ((const float4*)(in + ((size_t)b * NCO_ + chIn + c) * NT_ +
                              (size_t)n * T_))[t4];
#pragma unroll
  for (int o = 0; o < C_; ++o) {
    float bo = bias[o];
    float4 s = make_float4(bo, bo, bo, bo);
#pragma unroll
    for (int c = 0; c < C_; ++c) {
      float wc = w[o * C_ + c];
      s.x += wc * xin[c].x; s.y += wc * xin[c].y;
      s.z += wc * xin[c].z; s.w += wc * xin[c].w;
    }
    ((float4*)(x1p + ((size_t)(b * C_ + o) * NP_ + n) * T_))[t4] = s;
  }
}

// ---------------------------------------------------------------------------
// Kernel A2: xs[b,c,n] = sum_t x1p[b,c,n,t]  (float4 loads, LDS reduction)
// ---------------------------------------------------------------------------
__global__ __launch_bounds__(256)
void mh_xsum(const float* __restrict__ x1p, float* __restrict__ xs) {
  const int row = blockIdx.x;                    // over B*C*N_
  const int n = row % N_;
  const int bc = row / N_;
  const float4* p =
      (const float4*)(x1p + ((size_t)bc * NP_ + n) * T_);
  float4 v = p[threadIdx.x];                     // 256 * 4 = 1024 = T_
  float s = v.x + v.y + v.z + v.w;
  __shared__ float red[256];
  red[threadIdx.x] = s;
  __syncthreads();
  for (int st = 128; st > 0; st >>= 1) {
    if (threadIdx.x < st) red[threadIdx.x] += red[threadIdx.x + st];
    __syncthreads();
  }
  if (threadIdx.x == 0) xs[row] = red[0];
}

// ---------------------------------------------------------------------------
// Kernel B: adjacency row (b, n) -> adj[b, n, 0..61]
// ---------------------------------------------------------------------------
static __device__ float mh_block_softmax(float v, bool valid, float* red) {
  const int tid = threadIdx.x;
  red[tid] = valid ? v : -INFINITY;
  __syncthreads();
  for (int s = 32; s > 0; s >>= 1) {
    if (tid < s) red[tid] = fmaxf(red[tid], red[tid + s]);
    __syncthreads();
  }
  float mx = red[0];
  __syncthreads();
  float e = valid ? __expf(v - mx) : 0.f;
  red[tid] = e;
  __syncthreads();
  for (int s = 32; s > 0; s >>= 1) {
    if (tid < s) red[tid] += red[tid + s];
    __syncthreads();
  }
  float sum = red[0];
  __syncthreads();
  return valid ? (e / sum) : 0.f;
}

__global__ __launch_bounds__(64)
void mh_adj(const float* __restrict__ xs,      // [B, C, N]
            const float* __restrict__ mem,     // [C, N] (this layer)
            const float* __restrict__ fcw,     // 2 floats
            const float* __restrict__ fcb,     // 1 float
            float* __restrict__ adjOut) {      // [B, N, N] (this layer)
  const int n = blockIdx.x;
  const int b = blockIdx.y;
  const int m = threadIdx.x;          // 0..63, valid < 62
  const bool valid = (m < N_);
  __shared__ float red[64];
  __shared__ float afs[64];

  const float* xsB = xs + (size_t)b * C_ * N_;
  float xsn[C_];
#pragma unroll
  for (int c = 0; c < C_; ++c) xsn[c] = xsB[c * N_ + n];

  const float inv = 0.44721359549995793f;   // 1/sqrt(5)
  float v1 = 0.f, v2 = 0.f;
  if (valid) {
#pragma unroll
    for (int c = 0; c < C_; ++c) {
      v1 += xsn[c] * mem[c * N_ + m];
      v2 += xsn[c] * xsB[c * N_ + m];
    }
    v1 = fmaxf(v1 * inv, 0.f);
    v2 = fmaxf(v2 * inv, 0.f);
  }
  float a1 = mh_block_softmax(v1, valid, red);
  float a2 = mh_block_softmax(v2, valid, red);
  float af = fcw[0] * a1 + fcw[1] * a2 + fcb[0];
  af = mh_block_softmax(af, valid, red);

  // top-k (k=49) with jax.lax.top_k tie-breaking (lower index wins)
  afs[m] = valid ? af : -INFINITY;
  __syncthreads();
  if (valid) {
    int rank = 0;
    for (int j = 0; j < N_; ++j) {
      float aj = afs[j];
      rank += (aj > af) || (aj == af && j < m);
    }
    adjOut[((size_t)b * N_ + n) * N_ + m] = (rank < TOPK_) ? af : 0.f;
  }
}

// ---------------------------------------------------------------------------
// Kernel C: diffusion matmul on the matrix pipe + GCN 1x1 + skip, fused.
//   y[c,m,t]     = sum_n adj[b,n,m] * x1p[b,c,n,t]     (V_WMMA_F32_16X16X4)
//   out[b,o,m,t] = gb[o] + sum_c gw[o,c]*y[c,m,t] + skip[b,o,m,t]
// Block = 8 waves; one adj^T tile (16m x 64k, zero-padded) staged in LDS and
// shared by all waves; each wave owns one 16-wide t-slab. All inner-loop
// loads are unguarded (padding handles N=62 -> 64).
// ---------------------------------------------------------------------------
__global__ __launch_bounds__(256)
void mh_diffuse_wmma(const float* __restrict__ outR,  // d_out (skip read)
                     float* __restrict__ outW,        // d_out (layer write)
                     const float* __restrict__ x1p,   // [B, C, 64, T]
                     const float* __restrict__ adj,   // [B, N, N] this layer
                     const float* __restrict__ gw,    // 25
                     const float* __restrict__ gb,    // 5
                     int chIn, int chOut) {
  const int tid  = threadIdx.x;
  const int lane = tid & 31;
  const int wave = tid >> 5;          // 0..7
  const int mt = blockIdx.y;          // 0..3   -> m tile
  const int b  = blockIdx.z;          // 0..31
  const int mbase = mt * 16;
  const int tbase = (blockIdx.x * 8 + wave) * 16;
  const int row = lane & 15;
  const int hi  = lane >> 4;

  // ---- stage adj^T tile into LDS: ldsA[m][k], pitch 68, zero padded ----
  __shared__ float ldsA[16 * LDSP_];
  {
    const float* adjB = adj + (size_t)b * N_ * N_;
    for (int i = tid; i < 16 * 64; i += 256) {
      const int k  = i >> 4;          // 0..63
      const int mc = i & 15;          // 0..15 (consecutive lanes -> coalesced)
      const int mg = mbase + mc;
      ldsA[mc * LDSP_ + k] =
          (k < N_ && mg < N_) ? adjB[(size_t)k * N_ + mg] : 0.f;
    }
  }
  __syncthreads();

  const float* x1B = x1p + (size_t)b * C_ * NTP_;
#pragma unroll
  for (int c = 0; c < C_; ++c)        // gfx1250 prefetch of the x panels
    __builtin_prefetch(x1B + (size_t)c * NTP_ + (size_t)(2 * hi) * T_ +
                           tbase + row, 0, 1);

  v8f acc[C_];
#pragma unroll
  for (int c = 0; c < C_; ++c) acc[c] = (v8f)0.f;

  // K loop over n (64 incl. zero pad): 16 steps of K=4, no guards
  const float* xBase = x1B + (size_t)(2 * hi) * T_ + tbase + row;
#pragma unroll
  for (int kb = 0; kb < 16; ++kb) {
    const int k0 = kb * 4 + 2 * hi;
    v2f a = *(const v2f*)(ldsA + row * LDSP_ + k0);   // ds_load_b64
#pragma unroll
    for (int c = 0; c < C_; ++c) {
      const float* xc = xBase + (size_t)c * NTP_ + (size_t)(kb * 4) * T_;
      v2f bf;
      bf[0] = xc[0];
      bf[1] = xc[T_];
      acc[c] = __builtin_amdgcn_wmma_f32_16x16x4_f32(
          false, a, false, bf, (short)0, acc[c], false, false);
    }
  }

  // uniform GCN weights
  float w[C_ * C_], bias[C_];
#pragma unroll
  for (int i = 0; i < C_ * C_; ++i) w[i] = gw[i];
#pragma unroll
  for (int o = 0; o < C_; ++o) bias[o] = gb[o];

  const int t = tbase + row;
#pragma unroll
  for (int v = 0; v < 8; ++v) {
    const int mm = mbase + v + 8 * hi;      // D layout: VGPR v -> M=v / v+8
    if (mm >= N_) continue;
#pragma unroll
    for (int o = 0; o < C_; ++o) {
      float s = bias[o];
#pragma unroll
      for (int c = 0; c < C_; ++c) s += w[o * C_ + c] * acc[c][v];
      const size_t nt = (size_t)mm * T_ + t;
      const float skip = outR[((size_t)b * NCO_ + chIn  + o) * NT_ + nt];
      outW[((size_t)b * NCO_ + chOut + o) * NT_ + nt] = s + skip;
    }
  }
}

// ---------------------------------------------------------------------------
extern "C" void kernel_launch(void* const* d_in, const int* in_sizes, int n_in,
                              void* d_out, int out_size, void* d_ws, size_t ws_size,
                              hipStream_t stream) {
  const float* x       = (const float*)d_in[0];
  const float* conv1_w = (const float*)d_in[1];
  const float* conv1_b = (const float*)d_in[2];
  const float* memory  = (const float*)d_in[3];
  const float* fc_w    = (const float*)d_in[4];
  const float* fc_b    = (const float*)d_in[5];
  const float* gcn_w   = (const float*)d_in[6];
  const float* gcn_b   = (const float*)d_in[7];

  float* out   = (float*)d_out;
  float* ws_x1 = (float*)d_ws;                                   // [B,C,64,T]
  float* ws_xs = ws_x1 + (size_t)B_ * C_ * NTP_;                 // [B,C,N]

  const size_t adjOff = (size_t)B_ * NCO_ * NT_;                 // adj region

  // place x into output channel block 0 (unifies layer-input addressing)
  {
    size_t tot = (size_t)B_ * C_ * NT_ / 4;
    mh_copy_x<<<(tot + 255) / 256, 256, 0, stream>>>((const float4*)x, out);
  }

  for (int l = 0; l < L_; ++l) {
    const int chIn  = l * C_;
    const int chOut = (l + 1) * C_;
    float* adjL = out + adjOff + (size_t)l * B_ * N_ * N_;

    {
      size_t tot = (size_t)B_ * NP_ * (T_ / 4);
      mh_conv1<<<(tot + 255) / 256, 256, 0, stream>>>(
          out, chIn, conv1_w + l * C_ * C_, conv1_b + l * C_, ws_x1);
    }
    mh_xsum<<<B_ * C_ * N_, 256, 0, stream>>>(ws_x1, ws_xs);
    mh_adj<<<dim3(N_, B_), 64, 0, stream>>>(
        ws_xs, memory + l * C_ * N_, fc_w + l * 2, fc_b + l, adjL);
    mh_diffuse_wmma<<<dim3(T_ / 16 / 8, 4, B_), 256, 0, stream>>>(
        out, out, ws_x1, adjL, gcn_w + l * C_ * C_, gcn_b + l * C_,
        chIn, chOut);
  }
}